// Loss_18665927869012
// MI455X (gfx1250) — compile-verified
//
#include <hip/hip_runtime.h>

// YOLO-v1 loss for MI455X (gfx1250, wave32).
// Memory-bound: ~193 MB streamed, ~8.3 us roofline at 23.3 TB/s.
// Strategy: 1 thread per SxS cell, non-temporal b64 loads (zero reuse -> don't
// pollute the 192 MB L2), per-cell loss, wave32 reduction via
// V_WMMA_F32_16X16X4_F32 (A=partials, B=ones), deterministic block partials ->
// single-block finalize (no float atomics).

#define LAMBDA_COORD 5.0f
#define LAMBDA_NOOBJ 0.5f

typedef float v2f __attribute__((ext_vector_type(2)));
typedef float v8f __attribute__((ext_vector_type(8)));

__device__ __forceinline__ float sq(float x) { return x * x; }

// Faithful port of reference _iou (x/S, y/S, w/2, h/2; 0 when iw<=0 or ih<=0).
__device__ __forceinline__ float iou_fn(const float* a, const float* b) {
    float ax = a[0] / 7.0f, ay = a[1] / 7.0f, aw = a[2] * 0.5f, ah = a[3] * 0.5f;
    float bx = b[0] / 7.0f, by = b[1] / 7.0f, bw = b[2] * 0.5f, bh = b[3] * 0.5f;
    float axmin = ax - aw, axmax = ax + aw, aymin = ay - ah, aymax = ay + ah;
    float bxmin = bx - bw, bxmax = bx + bw, bymin = by - bh, bymax = by + bh;
    float iw = fminf(axmax, bxmax) - fmaxf(axmin, bxmin);
    float ih = fminf(aymax, bymax) - fmaxf(aymin, bymin);
    bool valid = (iw > 0.0f) && (ih > 0.0f);
    float inter  = iw * ih;
    float a_area = (axmax - axmin) * (aymax - aymin);
    float b_area = (bxmax - bxmin) * (bymax - bymin);
    float un = a_area + b_area - inter;
    return valid ? (inter / un) : 0.0f;
}

// Full wave32 sum. Uses V_WMMA_F32_16X16X4_F32 with B = ones:
// A[m][0]=local_m (lanes 0-15, VGPR0), A[m][2]=local_{m+16} (lanes 16-31, VGPR0),
// A[m][1]=A[m][3]=0  =>  D[m][n] = local_m + local_{m+16}.
// Sum of the 8 D VGPRs per lane = half-row-block sums; one shfl_xor(16) finishes.
// Requires EXEC all ones (callers keep every lane live).
__device__ __forceinline__ float wave_reduce_sum(float local) {
#if __has_builtin(__builtin_amdgcn_wmma_f32_16x16x4_f32)
    v2f A; A.x = local; A.y = 0.0f;
    v2f B; B.x = 1.0f;  B.y = 1.0f;
    v8f C = {0.f, 0.f, 0.f, 0.f, 0.f, 0.f, 0.f, 0.f};
    C = __builtin_amdgcn_wmma_f32_16x16x4_f32(
        /*neg_a=*/false, A, /*neg_b=*/false, B,
        /*c_mod=*/(short)0, C, /*reuse_a=*/false, /*reuse_b=*/false);
    float s = C[0] + C[1] + C[2] + C[3] + C[4] + C[5] + C[6] + C[7];
    s += __shfl_xor(s, 16, 32);
    return s;
#else
    #pragma unroll
    for (int off = 16; off > 0; off >>= 1) local += __shfl_xor(local, off, 32);
    return local;
#endif
}

__global__ __launch_bounds__(256)
void yolo_loss_partial(const float* __restrict__ pred,
                       const float* __restrict__ targ,
                       float* __restrict__ partial,
                       int n_cells) {
    int cell = blockIdx.x * 256 + threadIdx.x;
    // Keep every lane live (WMMA needs EXEC all-1s): clamp + mask instead of return.
    float live = (cell < n_cells) ? 1.0f : 0.0f;
    int c = (cell < n_cells) ? cell : (n_cells - 1);

    // 30 floats per cell, 8-byte aligned base (120 B stride) -> 15 x b64 NT loads.
    const v2f* pp = reinterpret_cast<const v2f*>(pred) + (size_t)c * 15;
    const v2f* tp = reinterpret_cast<const v2f*>(targ) + (size_t)c * 15;
    float p[30], t[30];
    #pragma unroll
    for (int i = 0; i < 15; ++i) {
        v2f v = __builtin_nontemporal_load(pp + i);
        v2f w = __builtin_nontemporal_load(tp + i);
        p[2 * i] = v.x; p[2 * i + 1] = v.y;
        t[2 * i] = w.x; t[2 * i + 1] = w.y;
    }

    float obj = (t[4] > 0.0f) ? 1.0f : 0.0f;

    // Responsible box: argmax over per-box IoU; first-max wins ties (box 0 iff iou0>=iou1).
    float iou0 = iou_fn(&p[0], &t[0]);
    float iou1 = iou_fn(&p[5], &t[5]);
    float r0 = (iou0 >= iou1) ? 1.0f : 0.0f;
    float r1 = 1.0f - r0;

    // xy loss vs responsible-masked targets (wh loss in the source compares a
    // tensor with itself -> identically zero; reproduced by omission).
    float xy = sq(p[0] - t[0] * r0) + sq(p[1] - t[1] * r0)
             + sq(p[5] - t[5] * r1) + sq(p[6] - t[6] * r1);
    float oc = sq(p[4] - t[4] * r0) + sq(p[9] - t[9] * r1);
    float nc = sq(p[4] - t[4]) + sq(p[9] - t[9]);
    float cls = 0.0f;
    #pragma unroll
    for (int i = 10; i < 30; ++i) cls += sq(p[i] - t[i]);

    float local = obj * (LAMBDA_COORD * xy + oc + cls)
                + (1.0f - obj) * LAMBDA_NOOBJ * nc;
    local *= live;

    float s = wave_reduce_sum(local);

    __shared__ float wsum[8];
    int lane = threadIdx.x & 31;
    int wave = threadIdx.x >> 5;
    if (lane == 0) wsum[wave] = s;
    __syncthreads();
    if (threadIdx.x == 0) {
        float b = 0.0f;
        #pragma unroll
        for (int i = 0; i < 8; ++i) b += wsum[i];   // fixed order: deterministic
        partial[blockIdx.x] = b;
    }
}

__global__ __launch_bounds__(256)
void yolo_loss_finalize(const float* __restrict__ partial,
                        int n_partials, float inv_n,
                        float* __restrict__ out) {
    float acc = 0.0f;
    for (int i = threadIdx.x; i < n_partials; i += 256) acc += partial[i];
    float s = wave_reduce_sum(acc);   // all 256 lanes live here

    __shared__ float wsum[8];
    int lane = threadIdx.x & 31;
    int wave = threadIdx.x >> 5;
    if (lane == 0) wsum[wave] = s;
    __syncthreads();
    if (threadIdx.x == 0) {
        float b = 0.0f;
        #pragma unroll
        for (int i = 0; i < 8; ++i) b += wsum[i];
        out[0] = b * inv_n;
    }
}

extern "C" void kernel_launch(void* const* d_in, const int* in_sizes, int n_in,
                              void* d_out, int out_size, void* d_ws, size_t ws_size,
                              hipStream_t stream) {
    const float* pred = (const float*)d_in[0];
    const float* targ = (const float*)d_in[1];
    float* out = (float*)d_out;
    float* partial = (float*)d_ws;

    int total   = in_sizes[0];       // BATCH * 7 * 7 * 30
    int n_cells = total / 30;        // 802816 for the reference shapes
    int blocks  = (n_cells + 255) / 256;   // 3136, exact fit
    int batch   = n_cells / 49;      // divide loss by N = BATCH

    yolo_loss_partial<<<blocks, 256, 0, stream>>>(pred, targ, partial, n_cells);
    yolo_loss_finalize<<<1, 256, 0, stream>>>(partial, blocks, 1.0f / (float)batch, out);
}